// graphNet_16415365005697
// MI455X (gfx1250) — compile-verified
//
#include <hip/hip_runtime.h>
#include <cmath>

typedef float v2f __attribute__((ext_vector_type(2)));
typedef float v8f __attribute__((ext_vector_type(8)));

// ---------------------------------------------------------------------------
// Padding / degree / norm kernels (32-bit indexing: all extents < 2^31)
// ---------------------------------------------------------------------------

__global__ void k_pad_x(const float* __restrict__ x, float* __restrict__ out,
                        unsigned n, unsigned npad, unsigned din, unsigned din_p) {
  unsigned tid = blockIdx.x * blockDim.x + threadIdx.x;
  unsigned total = npad * din_p;
  if (tid >= total) return;
  unsigned c = tid % din_p;
  unsigned v = tid / din_p;
  out[tid] = (v < n && c < din) ? x[v * din + c] : 0.0f;
}

__global__ void k_pad_w(const float* __restrict__ w, float* __restrict__ wp,
                        unsigned din, unsigned dout, unsigned din_p, unsigned dout_p) {
  unsigned tid = blockIdx.x * blockDim.x + threadIdx.x;
  unsigned total = din_p * dout_p;
  if (tid >= total) return;
  unsigned r = tid / dout_p, c = tid % dout_p;
  wp[tid] = (r < din && c < dout) ? w[r * dout + c] : 0.0f;
}

__global__ void k_deg_init(float* __restrict__ deg, unsigned npad) {
  unsigned v = blockIdx.x * blockDim.x + threadIdx.x;
  if (v < npad) deg[v] = 1.0f;   // self-loop contributes 1 to every node
}

__global__ void k_deg_edges(const int* __restrict__ dstv, float* __restrict__ deg, int e) {
  int t = blockIdx.x * blockDim.x + threadIdx.x;
  if (t < e) unsafeAtomicAdd(&deg[dstv[t]], 1.0f);
}

__global__ void k_dinv(float* __restrict__ deg, unsigned npad) {
  unsigned v = blockIdx.x * blockDim.x + threadIdx.x;
  if (v < npad) deg[v] = rsqrtf(deg[v]);   // deg >= 1 always (self-loop)
}

__global__ void k_norm(const int* __restrict__ src, const int* __restrict__ dst,
                       const float* __restrict__ dinv, float* __restrict__ norm, int e) {
  int t = blockIdx.x * blockDim.x + threadIdx.x;
  if (t < e) norm[t] = dinv[src[t]] * dinv[dst[t]];
}

// ---------------------------------------------------------------------------
// WMMA GEMM: C[M=Npad, N=dout_p] = A[M, K=DINP] * B[K, N]
// Templated on DINP so the K loop fully unrolls (KT = DINP/4 wmma ops).
// Each wave owns one 16-wide column tile (blockIdx.y), preloads ALL B
// fragments into VGPRs once, then marches down M tiles reusing them:
// one global_load_b64 (A frag) per v_wmma_f32_16x16x4_f32.
// VGPR layouts (ISA 7.12.2):
//   A 16x4 f32: lane = 16*half + m holds A[m][2*half], A[m][2*half+1]
//   B 4x16 f32: lane = 16*half + n holds B[2*half][n], B[2*half+1][n]
//   C 16x16 f32: acc[r] at lane (half, idx) is C[r + 8*half][idx]
// All branches are wave-uniform -> EXEC all-ones around WMMA.
// ---------------------------------------------------------------------------
template <int DINP>
__global__ void k_gemm_wmma(const float* __restrict__ A, const float* __restrict__ B,
                            float* __restrict__ C, int dout_p, int ntm) {
  constexpr int KT = DINP / 4;
  int tn = blockIdx.y;
  int wavesPerCol = gridDim.x * (blockDim.x >> 5);
  int w = blockIdx.x * (blockDim.x >> 5) + (threadIdx.x >> 5);
  int lane = threadIdx.x & 31;
  int half = lane >> 4, idx = lane & 15;

  // Preload all B fragments for this column tile (register-resident weights).
  v2f bf[KT];
  const float* bcol = B + (long long)(2 * half) * dout_p + tn * 16 + idx;
#pragma unroll
  for (int k = 0; k < KT; ++k) {
    bf[k].x = bcol[(long long)(4 * k)     * dout_p];
    bf[k].y = bcol[(long long)(4 * k + 1) * dout_p];
  }

  for (int tm = w; tm < ntm; tm += wavesPerCol) {   // wave-uniform loop
    const float* arow = A + (long long)(tm * 16 + idx) * DINP + 2 * half;
    v8f acc = {0.f, 0.f, 0.f, 0.f, 0.f, 0.f, 0.f, 0.f};
#pragma unroll
    for (int k = 0; k < KT; ++k) {
      v2f a;
      a.x = arow[4 * k];
      a.y = arow[4 * k + 1];
      acc = __builtin_amdgcn_wmma_f32_16x16x4_f32(false, a, false, bf[k],
                                                  (short)0, acc, false, false);
    }
    float* crow = C + (long long)(tm * 16 + 8 * half) * dout_p + tn * 16 + idx;
#pragma unroll
    for (int r = 0; r < 8; ++r) crow[(long long)r * dout_p] = acc[r];
  }
}

// ---------------------------------------------------------------------------
// Aggregation: out[v] = dinv[v]^2 * h[v]  (self loop, also zero-inits pads)
//              out[dst] += norm_e * h[src]  (warp per edge, lanes = channels)
// Feature matrices are L2-resident (<= 70 MB vs 192 MB L2), so the random
// gather/atomic-scatter stays on-chip.
// ---------------------------------------------------------------------------
__global__ void k_agg_init(const float* __restrict__ h, const float* __restrict__ dinv,
                           float* __restrict__ out, unsigned npad, unsigned dout_p) {
  unsigned tid = blockIdx.x * blockDim.x + threadIdx.x;
  unsigned total = npad * dout_p;
  if (tid >= total) return;
  unsigned v = tid / dout_p;
  float d = dinv[v];
  out[tid] = d * d * h[tid];
}

__global__ void k_agg_edges(const int* __restrict__ src, const int* __restrict__ dst,
                            const float* __restrict__ norm,
                            const float* __restrict__ h, float* __restrict__ out,
                            int e, int dout, int dout_p) {
  long long gt = (long long)blockIdx.x * blockDim.x + threadIdx.x;
  int gw = (int)(gt >> 5);
  int lane = threadIdx.x & 31;
  if (gw >= e) return;
  int s = src[gw], d = dst[gw];
  float w = norm[gw];
  const float* hs = h + (long long)s * dout_p;
  float* od = out + (long long)d * dout_p;
  for (int c = lane; c < dout; c += 32)
    unsafeAtomicAdd(&od[c], w * hs[c]);
}

__global__ void k_bias_relu(float* __restrict__ out, const float* __restrict__ b,
                            unsigned n, unsigned dout, unsigned dout_p) {
  unsigned tid = blockIdx.x * blockDim.x + threadIdx.x;
  unsigned total = n * dout;
  if (tid >= total) return;
  unsigned v = tid / dout;
  unsigned c = tid % dout;
  unsigned idx = v * dout_p + c;
  out[idx] = fmaxf(out[idx] + b[c], 0.0f);
}

// ---------------------------------------------------------------------------
// log_softmax over C=21 valid columns (stride = padded 32); writes
// p_z to d_out[0 .. N*21) and z to d_out[N*21 .. 2*N*21)
// ---------------------------------------------------------------------------
__global__ void k_logsoftmax(const float* __restrict__ z, float* __restrict__ out,
                             int n, int C, int stride) {
  int v = blockIdx.x * blockDim.x + threadIdx.x;
  if (v >= n) return;
  const float* zr = z + (long long)v * stride;
  float m = -INFINITY;
  for (int c = 0; c < C; ++c) m = fmaxf(m, zr[c]);
  float s = 0.0f;
  for (int c = 0; c < C; ++c) s += __expf(zr[c] - m);
  float lse = m + __logf(s);
  float* p  = out + (long long)v * C;
  float* zo = out + (long long)n * C + (long long)v * C;
  for (int c = 0; c < C; ++c) { p[c] = zr[c] - lse; zo[c] = zr[c]; }
}

// ---------------------------------------------------------------------------
// Host side
// ---------------------------------------------------------------------------
static inline int cdivll(long long a, long long b) { return (int)((a + b - 1) / b); }

extern "C" void kernel_launch(void* const* d_in, const int* in_sizes, int n_in,
                              void* d_out, int out_size, void* d_ws, size_t ws_size,
                              hipStream_t stream) {
  const float* x  = (const float*)d_in[0];
  const int*   ei = (const int*)d_in[1];
  const float* W[4]  = {(const float*)d_in[2], (const float*)d_in[4],
                        (const float*)d_in[6], (const float*)d_in[8]};
  const float* Bi[4] = {(const float*)d_in[3], (const float*)d_in[5],
                        (const float*)d_in[7], (const float*)d_in[9]};

  const int dims[5]  = {21, 168, 84, 42, 21};
  const int pdims[5] = {32, 176, 96, 48, 32};

  int N = in_sizes[0] / dims[0];
  int E = in_sizes[1] / 2;
  long long Npad = ((long long)N + 15) / 16 * 16;
  const int* src  = ei;
  const int* dstv = ei + E;

  // workspace layout (floats)
  float* ws = (float*)d_ws;
  size_t off = 0;
  float* buf0 = ws + off; off += (size_t)Npad * 176;
  float* buf1 = ws + off; off += (size_t)Npad * 176;
  float* norm = ws + off; off += (size_t)E;
  float* dinv = ws + off; off += (size_t)Npad;   // used as deg, then dinv
  float* Wp[4];
  for (int i = 0; i < 4; ++i) { Wp[i] = ws + off; off += (size_t)pdims[i] * pdims[i + 1]; }

  const int TB = 256;

  // degrees -> dinv -> per-edge norm
  k_deg_init <<<cdivll(Npad, TB), TB, 0, stream>>>(dinv, (unsigned)Npad);
  k_deg_edges<<<cdivll(E, TB),    TB, 0, stream>>>(dstv, dinv, E);
  k_dinv     <<<cdivll(Npad, TB), TB, 0, stream>>>(dinv, (unsigned)Npad);
  k_norm     <<<cdivll(E, TB),    TB, 0, stream>>>(src, dstv, dinv, norm, E);

  // zero-padded inputs
  k_pad_x<<<cdivll(Npad * pdims[0], TB), TB, 0, stream>>>(
      x, buf0, (unsigned)N, (unsigned)Npad, dims[0], pdims[0]);
  for (int i = 0; i < 4; ++i)
    k_pad_w<<<cdivll(pdims[i] * pdims[i + 1], TB), TB, 0, stream>>>(
        W[i], Wp[i], dims[i], dims[i + 1], pdims[i], pdims[i + 1]);

  // layers: GEMM (WMMA) -> self-loop init -> edge scatter -> bias+relu
  for (int l = 0; l < 4; ++l) {
    int din_p  = pdims[l];
    int dout   = dims[l + 1];
    int dout_p = pdims[l + 1];
    int ntn = dout_p / 16;
    int ntm = (int)(Npad / 16);

    dim3 ggrid(96, ntn);   // 96*8 = 768 waves per column tile
    switch (din_p) {
      case 32:  k_gemm_wmma<32> <<<ggrid, 256, 0, stream>>>(buf0, Wp[l], buf1, dout_p, ntm); break;
      case 176: k_gemm_wmma<176><<<ggrid, 256, 0, stream>>>(buf0, Wp[l], buf1, dout_p, ntm); break;
      case 96:  k_gemm_wmma<96> <<<ggrid, 256, 0, stream>>>(buf0, Wp[l], buf1, dout_p, ntm); break;
      case 48:  k_gemm_wmma<48> <<<ggrid, 256, 0, stream>>>(buf0, Wp[l], buf1, dout_p, ntm); break;
    }

    k_agg_init<<<cdivll(Npad * dout_p, TB), TB, 0, stream>>>(
        buf1, dinv, buf0, (unsigned)Npad, (unsigned)dout_p);
    k_agg_edges<<<cdivll((long long)E * 32, TB), TB, 0, stream>>>(
        src, dstv, norm, buf1, buf0, E, dout, dout_p);
    k_bias_relu<<<cdivll((long long)N * dout, TB), TB, 0, stream>>>(
        buf0, Bi[l], (unsigned)N, (unsigned)dout, (unsigned)dout_p);
  }

  k_logsoftmax<<<cdivll(N, TB), TB, 0, stream>>>(buf0, (float*)d_out, N, dims[4], pdims[4]);
}